// PartialStringGNNPerturbModel_6923487281944
// MI455X (gfx1250) — compile-verified
//
#include <hip/hip_runtime.h>
#include <hip/hip_bf16.h>
#include <math.h>

// ---------------------------------------------------------------------------
// Problem constants (from the reference)
// ---------------------------------------------------------------------------
#define NN   18870   // graph nodes
#define DD   256     // embedding dim
#define HH   512     // hidden
#define XPW  2048    // MLP expansion
#define RR   512     // per-class proj dim
#define GG   6640    // genes
#define CC   3       // classes
#define NBLK 6       // MLP blocks
#define EE   600000  // edges
#define BB   4096    // batch

typedef float v2f __attribute__((ext_vector_type(2)));
typedef float v8f __attribute__((ext_vector_type(8)));

// Tell InferAddressSpaces these generic pointers are global memory so loads
// lower to global_load_* (LOADcnt only) instead of flat_load_* (LOADcnt+DScnt).
__device__ __forceinline__ void assume_global(const void* p) {
    __builtin_assume(!__builtin_amdgcn_is_shared(p) && !__builtin_amdgcn_is_private(p));
}

// ---------------------------------------------------------------------------
// Small elementwise / scatter kernels
// ---------------------------------------------------------------------------
__global__ void k_fill0(float* __restrict__ p, int n) {
    int i = blockIdx.x * blockDim.x + threadIdx.x;
    if (i < n) p[i] = 0.0f;
}

__global__ void k_scatter_deg(const int* __restrict__ dst, const float* __restrict__ ew,
                              float* __restrict__ deg, int E) {
    int e = blockIdx.x * blockDim.x + threadIdx.x;
    if (e < E) atomicAdd(&deg[dst[e]], ew[e]);
}

__global__ void k_dinv(const float* __restrict__ deg, float* __restrict__ dinv, int n) {
    int i = blockIdx.x * blockDim.x + threadIdx.x;
    if (i < n) {
        float d = deg[i];
        dinv[i] = (d > 0.0f) ? rsqrtf(fmaxf(d, 1e-12f)) : 0.0f;
    }
}

__global__ void k_edge_norm(const int* __restrict__ src, const int* __restrict__ dst,
                            const float* __restrict__ ew, const float* __restrict__ dinv,
                            float* __restrict__ norm, int E) {
    int e = blockIdx.x * blockDim.x + threadIdx.x;
    if (e < E) norm[e] = dinv[src[e]] * ew[e] * dinv[dst[e]];
}

// acc[n,d] = x[n,d] + bias[d]   (folds GCN bias + residual into the scatter acc)
__global__ void k_init_acc(const float* __restrict__ x, const float* __restrict__ bias,
                           float* __restrict__ acc, int total) {
    int i = blockIdx.x * blockDim.x + threadIdx.x;
    if (i < total) acc[i] = x[i] + bias[i & (DD - 1)];
}

// acc[dst[e], :] += h[src[e], :] * norm[e]   (64 threads/edge, 4 floats each)
__global__ void k_scatter_msg(const int* __restrict__ src, const int* __restrict__ dst,
                              const float* __restrict__ norm, const float* __restrict__ h,
                              float* __restrict__ acc, int E) {
    long tid = (long)blockIdx.x * blockDim.x + threadIdx.x;
    int e  = (int)(tid >> 6);
    int d4 = (int)(tid & 63) << 2;
    if (e >= E) return;
    int   s  = src[e];
    int   t  = dst[e];
    float nw = norm[e];
    const float4 hv = *(const float4*)(h + (long)s * DD + d4);
    float* ap = acc + (long)t * DD + d4;
    atomicAdd(ap + 0, hv.x * nw);
    atomicAdd(ap + 1, hv.y * nw);
    atomicAdd(ap + 2, hv.z * nw);
    atomicAdd(ap + 3, hv.w * nw);
}

// emb[b,:] = (idx[b] >= N) ? oov : adapted[idx[b], :]
__global__ void k_gather_emb(const int* __restrict__ gidx, const float* __restrict__ adapted,
                             const float* __restrict__ oov, float* __restrict__ emb, int total) {
    int i = blockIdx.x * blockDim.x + threadIdx.x;
    if (i >= total) return;
    int b = i >> 8;          // /DD
    int d = i & (DD - 1);
    int idx = gidx[b];
    emb[i] = (idx >= NN) ? oov[d] : adapted[(long)idx * DD + d];
}

// LayerNorm over H=512 per row; 256 threads, 2 elements each.
__global__ void k_layernorm(const float* __restrict__ h, const float* __restrict__ w,
                            const float* __restrict__ b, float* __restrict__ out) {
    __shared__ float red[256];
    int row = blockIdx.x;
    int t   = threadIdx.x;
    const float* x = h + (long)row * HH;
    float v0 = x[t], v1 = x[t + 256];
    red[t] = v0 + v1;
    __syncthreads();
    for (int off = 128; off > 0; off >>= 1) {
        if (t < off) red[t] += red[t + off];
        __syncthreads();
    }
    float mu = red[0] * (1.0f / HH);
    __syncthreads();
    float d0 = v0 - mu, d1 = v1 - mu;
    red[t] = d0 * d0 + d1 * d1;
    __syncthreads();
    for (int off = 128; off > 0; off >>= 1) {
        if (t < off) red[t] += red[t + off];
        __syncthreads();
    }
    float rstd = rsqrtf(red[0] * (1.0f / HH) + 1e-5f);
    float* o = out + (long)row * HH;
    o[t]       = d0 * rstd * w[t]       + b[t];
    o[t + 256] = d1 * rstd * w[t + 256] + b[t + 256];
}

// ---------------------------------------------------------------------------
// WMMA fp32 GEMM:  C[M,Nn] = epilogue(A[M,K] @ B + bias)
//   B(k,n) = Bm[k*LDBK + n*LDBN]   (compile-time strides -> immediate-offset
//   loads and constant pointer bumps; LDBK=1,LDBN=R for the oge^T einsum)
//
// Wave tile = 32 rows x 64 cols = 2x4 WMMA subtiles; 8 wmma per K-step of 4.
// Ragged M / ragged N handled by CLAMPING source indices (a WMMA output
// row/col only depends on the same fragment row/col, so clamped garbage lands
// exclusively in unstored elements) -- no divergent loads around the WMMA.
//
// Fragment layouts per CDNA5 ISA 7.12.2 (32-bit, wave32):
//   A 16x4 : lanes 0-15 -> M rows, VGPR0={K0|K2 by lane-half}, VGPR1={K1|K3}
//   B 4x16 : lanes 0-15 -> N cols, same K split across lane halves
//   C 16x16: VGPR r -> M = m0 + r (+8 for upper lane half), N = n0 + lane%16
// ---------------------------------------------------------------------------
#define EPI_NONE 0
#define EPI_GELU 1
#define EPI_ADD  2

__device__ __forceinline__ float gelu_exact(float v) {
    return 0.5f * v * (1.0f + erff(v * 0.70710678118654752f));
}

template <int EPI, int LDBK, int LDBN>
__global__ void __launch_bounds__(256)
k_gemm_wmma_f32(const float* __restrict__ A, const float* __restrict__ Bm,
                const float* __restrict__ bias, float* __restrict__ C,
                int M, int Nn, int K) {
    assume_global(A);
    assume_global(Bm);
    assume_global(C);

    const int wave  = threadIdx.x >> 5;
    const int lane  = threadIdx.x & 31;
    const int n0    = (blockIdx.x * 8 + wave) * 64;
    if (n0 >= Nn) return;                 // wave-uniform exit; EXEC stays all-1
    const int m0 = blockIdx.y * 32;

    const int laneLo  = lane & 15;        // row within A frag / col within B frag
    const int laneHiK = (lane >> 4) << 1; // +0 or +2 K offset for this lane half

    // A fragment pointers for the two 16-row subtiles (rows clamped).
    int ar0 = min(m0 + laneLo,      M - 1);
    int ar1 = min(m0 + 16 + laneLo, M - 1);
    const float* ap0 = A + (long)ar0 * K + laneHiK;
    const float* ap1 = A + (long)ar1 * K + laneHiK;

    // B fragment pointers for the four 16-col subtiles (cols clamped).
    // Named scalars (not an array) so SROA keeps them in VGPRs and the
    // address-space inference isn't defeated.
    int bc0 = min(n0 +  0 + laneLo, Nn - 1);
    int bc1 = min(n0 + 16 + laneLo, Nn - 1);
    int bc2 = min(n0 + 32 + laneLo, Nn - 1);
    int bc3 = min(n0 + 48 + laneLo, Nn - 1);
    const long bko = (long)laneHiK * LDBK;
    const float* bp0 = Bm + (long)bc0 * LDBN + bko;
    const float* bp1 = Bm + (long)bc1 * LDBN + bko;
    const float* bp2 = Bm + (long)bc2 * LDBN + bko;
    const float* bp3 = Bm + (long)bc3 * LDBN + bko;

    v8f c00 = {}, c01 = {}, c02 = {}, c03 = {};
    v8f c10 = {}, c11 = {}, c12 = {}, c13 = {};
    for (int k0 = 0; k0 < K; k0 += 4) {
        v2f a0 = *(const v2f*)ap0;  ap0 += 4;
        v2f a1 = *(const v2f*)ap1;  ap1 += 4;
        v2f b0, b1, b2, b3;
        b0.x = bp0[0]; b0.y = bp0[LDBK]; bp0 += 4 * LDBK;
        b1.x = bp1[0]; b1.y = bp1[LDBK]; bp1 += 4 * LDBK;
        b2.x = bp2[0]; b2.y = bp2[LDBK]; bp2 += 4 * LDBK;
        b3.x = bp3[0]; b3.y = bp3[LDBK]; bp3 += 4 * LDBK;
        c00 = __builtin_amdgcn_wmma_f32_16x16x4_f32(false, a0, false, b0, (short)0, c00, false, false);
        c10 = __builtin_amdgcn_wmma_f32_16x16x4_f32(false, a1, false, b0, (short)0, c10, false, false);
        c01 = __builtin_amdgcn_wmma_f32_16x16x4_f32(false, a0, false, b1, (short)0, c01, false, false);
        c11 = __builtin_amdgcn_wmma_f32_16x16x4_f32(false, a1, false, b1, (short)0, c11, false, false);
        c02 = __builtin_amdgcn_wmma_f32_16x16x4_f32(false, a0, false, b2, (short)0, c02, false, false);
        c12 = __builtin_amdgcn_wmma_f32_16x16x4_f32(false, a1, false, b2, (short)0, c12, false, false);
        c03 = __builtin_amdgcn_wmma_f32_16x16x4_f32(false, a0, false, b3, (short)0, c03, false, false);
        c13 = __builtin_amdgcn_wmma_f32_16x16x4_f32(false, a1, false, b3, (short)0, c13, false, false);
    }

    // Epilogue: bias / GELU / residual-add, with row/col store guards.
    const int rb = (lane >> 4) << 3;      // +0 or +8 within a 16-row subtile
    const v8f* cs[8] = { &c00, &c10, &c01, &c11, &c02, &c12, &c03, &c13 };
#pragma unroll
    for (int j = 0; j < 4; ++j) {
        int col = n0 + j * 16 + laneLo;
        if (col >= Nn) continue;
        float bv = bias ? bias[col] : 0.0f;
#pragma unroll
        for (int mi = 0; mi < 2; ++mi) {
            const v8f& cc = *cs[j * 2 + mi];
#pragma unroll
            for (int r = 0; r < 8; ++r) {
                int row = m0 + mi * 16 + rb + r;
                if (row < M) {
                    float v = cc[r] + bv;
                    if (EPI == EPI_GELU) v = gelu_exact(v);
                    long idx = (long)row * Nn + col;
                    if (EPI == EPI_ADD) v += C[idx];
                    C[idx] = v;
                }
            }
        }
    }
}

// ---------------------------------------------------------------------------
// Orchestration
// ---------------------------------------------------------------------------
static inline int cdiv_i(long a, long b) { return (int)((a + b - 1) / b); }
static inline dim3 gemm_grid(int M, int Nn) {
    return dim3(cdiv_i(cdiv_i(Nn, 64), 8), cdiv_i(M, 32));
}

extern "C" void kernel_launch(void* const* d_in, const int* in_sizes, int n_in,
                              void* d_out, int out_size, void* d_ws, size_t ws_size,
                              hipStream_t stream) {
    (void)in_sizes; (void)n_in; (void)out_size; (void)ws_size;

    const int*   gidx   = (const int*)  d_in[0];
    const int*   esrc   = (const int*)  d_in[1];          // edge_index[0,:]
    const int*   edst   = ((const int*) d_in[1]) + EE;    // edge_index[1,:]
    const float* ew     = (const float*)d_in[2];
    const float* x0     = (const float*)d_in[3];
    const float* W6     = (const float*)d_in[4];
    const float* b6     = (const float*)d_in[5];
    const float* W7     = (const float*)d_in[6];
    const float* b7     = (const float*)d_in[7];
    const float* Wpost  = (const float*)d_in[8];
    const float* bpost  = (const float*)d_in[9];
    const float* oov    = (const float*)d_in[10];
    const float* Win    = (const float*)d_in[11];
    const float* bin    = (const float*)d_in[12];
    const float* lnw    = (const float*)d_in[13];
    const float* lnb    = (const float*)d_in[14];
    const float* W1     = (const float*)d_in[15];
    const float* b1     = (const float*)d_in[16];
    const float* W2     = (const float*)d_in[17];
    const float* b2     = (const float*)d_in[18];
    const float* Wout   = (const float*)d_in[19];
    const float* bout   = (const float*)d_in[20];
    const float* oge    = (const float*)d_in[21];
    float*       out    = (float*)d_out;

    // --- workspace carve ---
    float* ws = (float*)d_ws;
    size_t off = 0;
    auto alloc = [&](size_t n) { float* p = ws + off; off += n; return p; };
    float* deg   = alloc(NN);
    float* dinv  = alloc(NN);
    float* enorm = alloc(EE);
    float* hN    = alloc((size_t)NN * DD);  // GEMM result buffer (GCN layers)
    float* x1    = alloc((size_t)NN * DD);  // layer-1 out, later "adapted"
    float* x2    = alloc((size_t)NN * DD);  // layer-2 out
    float* emb   = alloc((size_t)BB * DD);
    float* hB    = alloc((size_t)BB * HH);
    float* zln   = alloc((size_t)BB * HH);
    float* z1    = alloc((size_t)BB * XPW);
    float* proj  = alloc((size_t)BB * CC * RR);

    const dim3 blk(256);

    // --- GCN normalization (shared by both layers) ---
    k_fill0<<<cdiv_i(NN, 256), 256, 0, stream>>>(deg, NN);
    k_scatter_deg<<<cdiv_i(EE, 256), 256, 0, stream>>>(edst, ew, deg, EE);
    k_dinv<<<cdiv_i(NN, 256), 256, 0, stream>>>(deg, dinv, NN);
    k_edge_norm<<<cdiv_i(EE, 256), 256, 0, stream>>>(esrc, edst, ew, dinv, enorm, EE);

    // --- GCN layer 1: x1 = scatter(x0@W6) + b6 + x0 ---
    k_gemm_wmma_f32<EPI_NONE, DD, 1><<<gemm_grid(NN, DD), blk, 0, stream>>>(
        x0, W6, nullptr, hN, NN, DD, DD);
    k_init_acc<<<cdiv_i((long)NN * DD, 256), 256, 0, stream>>>(x0, b6, x1, NN * DD);
    k_scatter_msg<<<cdiv_i((long)EE * 64, 256), 256, 0, stream>>>(esrc, edst, enorm, hN, x1, EE);

    // --- GCN layer 2: x2 = scatter(x1@W7) + b7 + x1 ---
    k_gemm_wmma_f32<EPI_NONE, DD, 1><<<gemm_grid(NN, DD), blk, 0, stream>>>(
        x1, W7, nullptr, hN, NN, DD, DD);
    k_init_acc<<<cdiv_i((long)NN * DD, 256), 256, 0, stream>>>(x1, b7, x2, NN * DD);
    k_scatter_msg<<<cdiv_i((long)EE * 64, 256), 256, 0, stream>>>(esrc, edst, enorm, hN, x2, EE);

    // --- adapted = x2 @ W_post + b_post  (reuse x1) ---
    k_gemm_wmma_f32<EPI_NONE, DD, 1><<<gemm_grid(NN, DD), blk, 0, stream>>>(
        x2, Wpost, bpost, x1, NN, DD, DD);

    // --- gather with OOV ---
    k_gather_emb<<<cdiv_i((long)BB * DD, 256), 256, 0, stream>>>(gidx, x1, oov, emb, BB * DD);

    // --- h = emb @ W_in + b_in ---
    k_gemm_wmma_f32<EPI_NONE, HH, 1><<<gemm_grid(BB, HH), blk, 0, stream>>>(
        emb, Win, bin, hB, BB, HH, DD);

    // --- 6 residual MLP blocks ---
    for (int i = 0; i < NBLK; ++i) {
        k_layernorm<<<BB, 256, 0, stream>>>(hB, lnw + (size_t)i * HH, lnb + (size_t)i * HH, zln);
        k_gemm_wmma_f32<EPI_GELU, XPW, 1><<<gemm_grid(BB, XPW), blk, 0, stream>>>(
            zln, W1 + (size_t)i * HH * XPW, b1 + (size_t)i * XPW, z1, BB, XPW, HH);
        k_gemm_wmma_f32<EPI_ADD, HH, 1><<<gemm_grid(BB, HH), blk, 0, stream>>>(
            z1, W2 + (size_t)i * XPW * HH, b2 + (size_t)i * HH, hB, BB, HH, XPW);
    }

    // --- proj = h @ W_out + b_out  -> [B, C*R] ---
    k_gemm_wmma_f32<EPI_NONE, CC * RR, 1><<<gemm_grid(BB, CC * RR), blk, 0, stream>>>(
        hB, Wout, bout, proj, BB, CC * RR, HH);

    // --- logits[b,c,g] = sum_r proj[(b*3+c), r] * oge[g, r] ---
    // B-matrix is oge^T accessed via compile-time strides (LDBK=1, LDBN=R).
    k_gemm_wmma_f32<EPI_NONE, 1, RR><<<gemm_grid(BB * CC, GG), blk, 0, stream>>>(
        proj, oge, nullptr, out, BB * CC, GG, RR);
}